// OS2RiRoIAlign_489626272159
// MI455X (gfx1250) — compile-verified
//
#include <hip/hip_runtime.h>
#include <hip/hip_bf16.h>
#include <math.h>

typedef __attribute__((ext_vector_type(2))) float v2f;
typedef __attribute__((ext_vector_type(8))) float v8f;

#define OUT_SIZE 7
#define N_ORI 8
#define SPATIAL_SCALE 0.125f
#define C_TOT 256
#define Hc 128
#define Wc 128
#define P_GRID 14      // OUT_SIZE * SAMPLE_NUM
#define NPTS 196       // P_GRID^2
#define NBINS 49       // OUT_SIZE^2
#define TILES 98       // (NBINS * 32) / 16 row-tiles for WMMA mixing

__global__ __launch_bounds__(256)
void os2r_roialign_kernel(const float* __restrict__ features,
                          const float* __restrict__ rois,
                          float* __restrict__ out) {
    __shared__ int   s_off[NPTS][4];          // corner offsets within one HxW plane
    __shared__ float s_w[NPTS][4];            // bilinear weights * 0.25 (sample avg)
    __shared__ float s_pool[NBINS * C_TOT];   // pooled[bin][channel], 49 KB

    const int roi = blockIdx.x;
    const int tid = threadIdx.x;

    // ---- RoI parameters (uniform across block) ----
    const float r0 = rois[roi * 6 + 0];
    const float cx = rois[roi * 6 + 1] * SPATIAL_SCALE;
    const float cy = rois[roi * 6 + 2] * SPATIAL_SCALE;
    const float rw = fmaxf(rois[roi * 6 + 3] * SPATIAL_SCALE, 1.0f);
    const float rh = fmaxf(rois[roi * 6 + 4] * SPATIAL_SCALE, 1.0f);
    const float th = rois[roi * 6 + 5];
    const int   b  = (int)r0;
    const float bin_h = rh / (float)OUT_SIZE;
    const float bin_w = rw / (float)OUT_SIZE;
    const float cth = cosf(th);
    const float sth = sinf(th);

    // ---- Stage 1: per-sample-point offsets + bilinear weights (196 pts) ----
    if (tid < NPTS) {
        const int py = tid / P_GRID;
        const int px = tid - py * P_GRID;
        const float fy = (float)(py >> 1) + ((py & 1) ? 0.75f : 0.25f);
        const float fx = (float)(px >> 1) + ((px & 1) ? 0.75f : 0.25f);
        const float yy = -0.5f * rh + fy * bin_h;
        const float xx = -0.5f * rw + fx * bin_w;
        float x = xx * cth - yy * sth + cx;
        float y = xx * sth + yy * cth + cy;
        const bool valid = (y >= -1.0f) && (y <= (float)Hc) &&
                           (x >= -1.0f) && (x <= (float)Wc);
        y = fminf(fmaxf(y, 0.0f), (float)(Hc - 1));
        x = fminf(fmaxf(x, 0.0f), (float)(Wc - 1));
        int y0 = min((int)floorf(y), Hc - 1);
        int x0 = min((int)floorf(x), Wc - 1);
        const int y1 = min(y0 + 1, Hc - 1);
        const int x1 = min(x0 + 1, Wc - 1);
        const float ly = y - (float)y0, lx = x - (float)x0;
        const float hy = 1.0f - ly,     hx = 1.0f - lx;
        const float s = valid ? 0.25f : 0.0f;   // fold 1/(SAMPLE_NUM^2)
        s_off[tid][0] = y0 * Wc + x0;  s_w[tid][0] = hy * hx * s;
        s_off[tid][1] = y0 * Wc + x1;  s_w[tid][1] = hy * lx * s;
        s_off[tid][2] = y1 * Wc + x0;  s_w[tid][2] = ly * hx * s;
        s_off[tid][3] = y1 * Wc + x1;  s_w[tid][3] = ly * lx * s;
    }
    __syncthreads();

    // ---- Stage 2: gather + 2x2 avg pool. One channel per thread. ----
    {
        const float* fc = features + ((size_t)(b * C_TOT + tid)) * (size_t)(Hc * Wc);
        for (int bin = 0; bin < NBINS; ++bin) {
            const int oy = bin / OUT_SIZE;
            const int ox = bin - oy * OUT_SIZE;
            float acc = 0.0f;
            #pragma unroll
            for (int sy = 0; sy < 2; ++sy) {
                #pragma unroll
                for (int sx = 0; sx < 2; ++sx) {
                    const int p = (oy * 2 + sy) * P_GRID + (ox * 2 + sx);
                    acc = fmaf(s_w[p][0], fc[s_off[p][0]], acc);
                    acc = fmaf(s_w[p][1], fc[s_off[p][1]], acc);
                    acc = fmaf(s_w[p][2], fc[s_off[p][2]], acc);
                    acc = fmaf(s_w[p][3], fc[s_off[p][3]], acc);
                }
            }
            s_pool[bin * C_TOT + tid] = acc;
        }
    }
    __syncthreads();

    // ---- Stage 3: orientation mixing via V_WMMA_F32_16X16X4_F32 ----
    // D[1568 x 8] = pooled[1568 x 8(orient)] x M[8 x 8]; 98 row-tiles, K=8 (2 WMMAs).
    {
        const int lane  = tid & 31;
        const int wave  = tid >> 5;
        const int n     = lane & 15;      // B/D column (orientation out)
        const int khalf = lane >> 4;      // A/B: lanes 16-31 hold K=2,3 (then +4)

        const float indf  = th * ((float)N_ORI / (2.0f * 3.14159265358979323846f));
        const float ind0f = floorf(indf);
        const float lv = indf - ind0f;
        const float rv = 1.0f - lv;
        int ind0 = (int)ind0f;
        ind0 = ((ind0 % N_ORI) + N_ORI) % N_ORI;

        const int kr = (n - ind0) & 7;    // two's-complement mask == mod 8
        const int kl = (kr + 1) & 7;
        const float cmask = (n < N_ORI) ? 1.0f : 0.0f;
        auto bval = [&](int k) -> float {
            float v = 0.0f;
            if (k == kr) v += rv;
            if (k == kl) v += lv;
            return v * cmask;
        };
        v2f B0, B1;
        B0.x = bval(khalf * 2 + 0);  B0.y = bval(khalf * 2 + 1);
        B1.x = bval(khalf * 2 + 4);  B1.y = bval(khalf * 2 + 5);

        for (int t = wave; t < TILES; t += 8) {         // wave-uniform trip count
            const int m   = t * 16 + n;                 // global row (bin, cgroup)
            const int bin = m >> 5;
            const int cb  = m & 31;
            const float* pp = &s_pool[bin * C_TOT + cb * N_ORI + khalf * 2];
            v2f A0; A0.x = pp[0]; A0.y = pp[1];         // K = 0..3 across lane halves
            v2f A1; A1.x = pp[4]; A1.y = pp[5];         // K = 4..7
            v8f d = {};
            d = __builtin_amdgcn_wmma_f32_16x16x4_f32(false, A0, false, B0,
                                                      (short)0, d, false, false);
            d = __builtin_amdgcn_wmma_f32_16x16x4_f32(false, A1, false, B1,
                                                      (short)0, d, false, false);
            if (n < N_ORI) {                            // only 8 valid columns
                const int rowoff = khalf << 3;          // lanes>=16 hold M+8
                #pragma unroll
                for (int vr = 0; vr < 8; ++vr) {
                    const int gm = t * 16 + vr + rowoff;
                    const int bo = gm >> 5;
                    const int co = (gm & 31) * N_ORI + n;
                    out[((size_t)roi * C_TOT + co) * NBINS + bo] = d[vr];
                }
            }
        }
    }
}

extern "C" void kernel_launch(void* const* d_in, const int* in_sizes, int n_in,
                              void* d_out, int out_size, void* d_ws, size_t ws_size,
                              hipStream_t stream) {
    const float* features = (const float*)d_in[0];
    const float* rois     = (const float*)d_in[1];
    float*       out      = (float*)d_out;
    const int K = in_sizes[1] / 6;   // rois is [K, 6]
    os2r_roialign_kernel<<<dim3(K), dim3(256), 0, stream>>>(features, rois, out);
}